// MMDLayer_6356551598365
// MI455X (gfx1250) — compile-verified
//
#include <hip/hip_runtime.h>

typedef __attribute__((ext_vector_type(2))) float v2f;
typedef __attribute__((ext_vector_type(8))) float v8f;

#define T_SEQ 2048
#define S_ST  512
#define D_LAT 128

// One wave computes one 16x16 tile of the banded Gram matrix G = Z * E^T via
// v_wmma_f32_16x16x4_f32 (32 K-steps over D=128), then converts it to
// (1 - (||z||^2 + ||e||^2 - 2G)/128)^3 and scatters into the shifted output band.
__global__ __launch_bounds__(128) void mmd_band_kernel(
    const float* __restrict__ zx,    // T x D   (x[0,:,0,:])
    const float* __restrict__ init,  // S x D
    float* __restrict__ out)         // T x S
{
    const int lane = threadIdx.x & 31;
    const int wave = threadIdx.x >> 5;   // 0..3, uniform within a wave
    const int m    = lane & 15;          // A: row index / B: col index
    const int kh   = lane >> 4;          // K-half select (K = 2*kh + {0,1})
    const int t0   = blockIdx.x * 16;    // row tile base

    // ---- Preload A fragments: rows z[t0..t0+15], all 128 K values -------------
    // Lane holds row (t0+m), elements d = 4*step + 2*kh + {0,1}  -> b64 loads.
    const float* zrow = zx + (size_t)(t0 + m) * D_LAT + 2 * kh;
    v2f a[32];
#pragma unroll
    for (int st = 0; st < 32; ++st)
        a[st] = *(const v2f*)(zrow + 4 * st);

    // Row norms: each lane sums its half of row (t0+m); xor-16 combines halves.
    float nzp = 0.f;
#pragma unroll
    for (int st = 0; st < 32; ++st)
        nzp += a[st].x * a[st].x + a[st].y * a[st].y;
    const float nzfull = nzp + __shfl_xor(nzp, 16, 32);
    // Accumulator VGPR r holds output row (t0 + r + 8*kh): broadcast its norm.
    float nz[8];
#pragma unroll
    for (int r = 0; r < 8; ++r)
        nz[r] = __shfl(nzfull, r + 8 * kh, 32);

    // ---- Band tiles: k in [t0, t0+527] covers window of all 16 rows ----------
    for (int kt = wave; kt < 33; kt += 4) {
        const int k0 = t0 + 16 * kt;
        const int k  = k0 + m;  // this lane's window column (N = m)
        // E[k] = reversed initial_state for k<S, else z[k-S]  (select, no EXEC change)
        const float* erow = (k < S_ST)
            ? (init + (size_t)(S_ST - 1 - k) * D_LAT + 2 * kh)
            : (zx   + (size_t)(k - S_ST)     * D_LAT + 2 * kh);

        v8f  c   = {0.f, 0.f, 0.f, 0.f, 0.f, 0.f, 0.f, 0.f};
        float nep = 0.f;
#pragma unroll
        for (int st = 0; st < 32; ++st) {
            v2f b = *(const v2f*)(erow + 4 * st);
            nep += b.x * b.x + b.y * b.y;   // column norm for free while loading
            // D = A(16x4) x B(4x16) + C : full fp32 matrix op
            c = __builtin_amdgcn_wmma_f32_16x16x4_f32(
                    false, a[st], false, b, (short)0, c, false, false);
        }
        const float ne = nep + __shfl_xor(nep, 16, 32);  // ||E_k||^2, k = k0+m

        // C layout: VGPR r, lane -> (M = r + 8*kh, N = m). Map to out[t][s].
#pragma unroll
        for (int r = 0; r < 8; ++r) {
            const int t = t0 + r + 8 * kh;
            const int s = S_ST + t - 1 - k;          // shift-buffer index
            if (s >= 0 && s < S_ST) {                // divergent only at stores
                const float dist = nz[r] + ne - 2.f * c[r];
                const float val  = 1.f - dist * (1.f / (float)D_LAT);
                out[(size_t)t * S_ST + s] = val * val * val;
            }
        }
    }
}

extern "C" void kernel_launch(void* const* d_in, const int* in_sizes, int n_in,
                              void* d_out, int out_size, void* d_ws, size_t ws_size,
                              hipStream_t stream) {
    const float* x    = (const float*)d_in[0];  // (1, T, 1, D) contiguous = T x D
    const float* init = (const float*)d_in[1];  // (S, D)
    float* out = (float*)d_out;                 // (1, T, S)
    (void)in_sizes; (void)n_in; (void)out_size; (void)d_ws; (void)ws_size;

    dim3 grid(T_SEQ / 16);   // 128 row tiles
    dim3 block(128);         // 4 waves; wave w handles k-tiles w, w+4, ...
    hipLaunchKernelGGL(mmd_band_kernel, grid, block, 0, stream, x, init, out);
}